// IGMambaModule_Symmetric_77300821394072
// MI455X (gfx1250) — compile-verified
//
#include <hip/hip_runtime.h>
#include <cstddef>
#include <cstdint>

#define B_ 2
#define C_ 96
#define H_ 96
#define W_ 96
#define HW_ (H_*W_)
#define L_ HW_
#define M_ (B_*L_)          // 18432 rows (b*L)
#define DI_ 192
#define DS_ 16
#define DTR_ 6
#define RI_ 24
#define PW_ 98              // padded conv width/height
#define PHW_ (PW_*PW_)
#define KC_ (C_*9)          // 864 conv reduction depth
#define OUTN_ ((size_t)B_*C_*HW_)   // 1769472 per output tensor

typedef float v2f __attribute__((ext_vector_type(2)));
typedef float v8f __attribute__((ext_vector_type(8)));

__device__ __forceinline__ float sigm_(float x){ return 1.f/(1.f+expf(-x)); }
__device__ __forceinline__ float lrelu_(float x){ return (x>=0.f)?x:0.1f*x; }

// async global->LDS copy of one element (gfx1250 GLOBAL_LOAD_ASYNC_TO_LDS_B32).
// LDS write conveyed via "memory" clobber; ordering via s_wait_asynccnt 0 +
// __syncthreads() before any ds_load of the staged data.
__device__ __forceinline__ void async_g2l_b32(float* lds_elem, const float* gsrc){
  unsigned lds = (unsigned)(uintptr_t)lds_elem;            // low 32b = LDS byte addr
  unsigned long long ga = (unsigned long long)(uintptr_t)gsrc;
  asm volatile("global_load_async_to_lds_b32 %0, %1, off"
               :: "v"(lds), "v"(ga) : "memory");
}
__device__ __forceinline__ void async_wait0(){
  asm volatile("s_wait_asynccnt 0" ::: "memory");
}

// ---------------- bilinear 2x upsample (48x48 -> 96x96, half-pixel centers) --------
__global__ void k_upsample2x(const float* __restrict__ src, float* __restrict__ dst){
  int i = blockIdx.x*blockDim.x + threadIdx.x;
  if (i >= B_*HW_) return;
  int b = i / HW_; int p = i % HW_; int h = p / W_; int w = p % W_;
  float sh = fminf(fmaxf(h*0.5f - 0.25f, 0.f), 47.f);
  float sw = fminf(fmaxf(w*0.5f - 0.25f, 0.f), 47.f);
  int h0 = (int)sh, w0 = (int)sw;
  int h1 = min(h0+1,47), w1 = min(w0+1,47);
  float fh = sh-h0, fw = sw-w0;
  const float* s = src + (size_t)b*48*48;
  float v = (1.f-fh)*((1.f-fw)*s[h0*48+w0] + fw*s[h0*48+w1])
          +      fh *((1.f-fw)*s[h1*48+w0] + fw*s[h1*48+w1]);
  dst[i] = v;
}

// ---------------- illumination mask (1x1 conv over 97ch) + f_low = x*m -------------
__global__ void k_mask_flow(const float* __restrict__ x, const float* __restrict__ ilu,
                            const float* __restrict__ icw, const float* __restrict__ icb,
                            float* __restrict__ flow){
  int i = blockIdx.x*blockDim.x + threadIdx.x;
  if (i >= B_*HW_) return;
  int b = i/HW_, p = i%HW_;
  const float* xb = x + (size_t)b*C_*HW_ + p;
  float s = icb[0] + ilu[i]*icw[C_];
  for (int c=0;c<C_;++c) s += xb[(size_t)c*HW_]*icw[c];
  float m = sigm_(s);
  float* fb = flow + (size_t)b*C_*HW_ + p;
  for (int c=0;c<C_;++c) fb[(size_t)c*HW_] = xb[(size_t)c*HW_]*m;
}

// ---------------- 3x3 box filter /9 + NCHW -> (B*L, C) transpose ------------------
__global__ void k_box_seq(const float* __restrict__ flow, float* __restrict__ seq){
  int i = blockIdx.x*blockDim.x + threadIdx.x;
  if (i >= B_*C_*HW_) return;
  int b = i/(C_*HW_); int r = i%(C_*HW_); int c = r/HW_; int p = r%HW_;
  int h = p/W_, w = p%W_;
  const float* f = flow + ((size_t)b*C_+c)*HW_;
  float s = 0.f;
  for (int dy=-1;dy<=1;++dy) for (int dx=-1;dx<=1;++dx){
    int hh=h+dy, ww=w+dx;
    if (hh>=0 && hh<H_ && ww>=0 && ww<W_) s += f[hh*W_+ww];
  }
  seq[((size_t)b*L_+p)*C_ + c] = s*(1.f/9.f);
}

// ---------------- row layernorm over C=96 -----------------------------------------
__global__ void k_layernorm(const float* __restrict__ in, const float* __restrict__ g,
                            const float* __restrict__ bt, float* __restrict__ out){
  int m = blockIdx.x*blockDim.x + threadIdx.x;
  if (m >= M_) return;
  const float* r = in + (size_t)m*C_;
  float s=0.f, s2=0.f;
  for (int c=0;c<C_;++c){ float v=r[c]; s+=v; s2+=v*v; }
  float mu = s*(1.f/C_); float var = s2*(1.f/C_) - mu*mu;
  float inv = rsqrtf(var + 1e-5f);
  float* o = out + (size_t)m*C_;
  for (int c=0;c<C_;++c) o[c] = (r[c]-mu)*inv*g[c] + bt[c];
}

// layernorm + transpose to NCHW (f_illum)
__global__ void k_layernorm_t(const float* __restrict__ in, const float* __restrict__ g,
                              const float* __restrict__ bt, float* __restrict__ out_nchw){
  int m = blockIdx.x*blockDim.x + threadIdx.x;
  if (m >= M_) return;
  int b = m/L_, p = m%L_;
  const float* r = in + (size_t)m*C_;
  float s=0.f, s2=0.f;
  for (int c=0;c<C_;++c){ float v=r[c]; s+=v; s2+=v*v; }
  float mu = s*(1.f/C_); float inv = rsqrtf(s2*(1.f/C_) - mu*mu + 1e-5f);
  for (int c=0;c<C_;++c)
    out_nchw[((size_t)b*C_+c)*HW_ + p] = (r[c]-mu)*inv*g[c] + bt[c];
}

// ---------------- zero-pad x_proj weights (38,192) -> (48,192) ---------------------
__global__ void k_padw(const float* __restrict__ w, float* __restrict__ wp){
  int i = blockIdx.x*blockDim.x + threadIdx.x;
  if (i >= 48*DI_) return;
  wp[i] = (i < 38*DI_) ? w[i] : 0.f;
}

// ---------------- WMMA f32 GEMM with async-LDS weight panel ------------------------
// Block = 8 waves = 8 consecutive M tiles sharing one N tile. W panel (16 x K) is
// DMA'd global->LDS once per block; waves read B fragments from LDS (ds_load_b64).
template<int K>
__global__ __launch_bounds__(256) void k_wmma_gemm_lds(
    const float* __restrict__ A, const float* __restrict__ Wt,
    float* __restrict__ Cc, int tiles_n, int Nld){
  __shared__ float wsh[16*K];
  int tn  = blockIdx.x % tiles_n;
  int tmb = (blockIdx.x / tiles_n) << 3;
  int tid = threadIdx.x;
  const float* wpanel = Wt + (size_t)(tn<<4)*K;
  for (int e = tid; e < 16*K; e += 256)
    async_g2l_b32(&wsh[e], wpanel + e);
  async_wait0();
  __syncthreads();

  int wave = tid >> 5; int lane = tid & 31;
  int tm = tmb + wave;
  int row  = (tm<<4) + (lane & 15);
  int koff = (lane < 16) ? 0 : 2;
  const v2f* Ar = (const v2f*)(A + (size_t)row*K + koff);
  const v2f* Wr = (const v2f*)(wsh + (lane&15)*K + koff);
  v8f acc = {};
  #pragma unroll 4
  for (int kk = 0; kk < K/4; ++kk){
    v2f av = Ar[2*kk];
    v2f bv = Wr[2*kk];
    acc = __builtin_amdgcn_wmma_f32_16x16x4_f32(false, av, false, bv,
                                                (short)0, acc, false, false);
  }
  int r0 = (tm<<4) + ((lane<16)?0:8);
  float* Co = Cc + (size_t)r0*Nld + (tn<<4) + (lane&15);
  #pragma unroll
  for (int i=0;i<8;++i) Co[(size_t)i*Nld] = acc[i];
}

// ---------------- padded pixel-major conv inputs -----------------------------------
// f_high = x*(1+alpha*grad_up) written directly as padded pixel-major (B,98,98,96)
__global__ void k_pad_fhigh_pm(const float* __restrict__ x, const float* __restrict__ gru,
                               const float* __restrict__ alpha, float* __restrict__ dst){
  int i = blockIdx.x*blockDim.x + threadIdx.x;
  if (i >= B_*PHW_*C_) return;
  int b = i/(PHW_*C_); int r = i%(PHW_*C_); int q = r/C_; int c = r%C_;
  int hh = q/PW_, ww = q%PW_;
  float v = 0.f;
  if (hh>=1 && hh<=H_ && ww>=1 && ww<=W_){
    int p = (hh-1)*W_ + (ww-1);
    v = x[((size_t)b*C_+c)*HW_ + p] * (1.f + alpha[0]*gru[b*HW_+p]);
  }
  dst[i] = v;
}
// pixel-major (M,96) -> padded pixel-major (B,98,98,96)
__global__ void k_pad_pm(const float* __restrict__ src, float* __restrict__ dst){
  int i = blockIdx.x*blockDim.x + threadIdx.x;
  if (i >= B_*PHW_*C_) return;
  int b = i/(PHW_*C_); int r = i%(PHW_*C_); int q = r/C_; int c = r%C_;
  int hh = q/PW_, ww = q%PW_;
  float v = 0.f;
  if (hh>=1 && hh<=H_ && ww>=1 && ww<=W_)
    v = src[((size_t)b*HW_ + (size_t)(hh-1)*W_ + (ww-1))*C_ + c];
  dst[i] = v;
}

// ---------------- WMMA implicit-GEMM 3x3 conv (96->96), pixel-major padded input ---
// Reduction order permuted to k = t*96 + ci (tap outer, channel inner); weights are
// permuted to match while async-DMA staging into LDS. Inner loop: one global b64
// load (affine address), one ds_load_b64, one v_wmma — no div/mod, no branches.
__global__ __launch_bounds__(256) void k_wmma_conv3x3_pm(
    const float* __restrict__ Xp, const float* __restrict__ Wt,
    float* __restrict__ Y){
  __shared__ float wsh[16*KC_];
  int tn  = blockIdx.x % 6;
  int tmb = (blockIdx.x / 6) << 3;
  int tid = threadIdx.x;
  const float* wpanel = Wt + (size_t)(tn<<4)*KC_;
  // stage permuted: wsh[n*864 + t*96 + ci] = W[n][ci][t]
  for (int e = tid; e < 16*KC_; e += 256){
    int n = e / KC_; int q = e - n*KC_; int t = q / C_; int ci = q - t*C_;
    async_g2l_b32(&wsh[e], wpanel + n*KC_ + ci*9 + t);
  }
  async_wait0();
  __syncthreads();

  int wave = tid >> 5; int lane = tid & 31;
  int tm = tmb + wave;
  int m = (tm<<4) + (lane&15);
  int b = m / HW_; int p = m % HW_; int h = p/W_; int w = p%W_;
  int koff = (lane<16)?0:2;
  const float* wrow = wsh + (lane&15)*KC_ + koff;
  v8f acc = {};
  #pragma unroll
  for (int t = 0; t < 9; ++t){
    const int ky = t/3, kx = t - (t/3)*3;   // compile-time after unroll
    const float* Abase = Xp + ((size_t)b*PHW_ + (size_t)(h+ky)*PW_ + (w+kx))*C_ + koff;
    const float* Brow  = wrow + t*C_;
    #pragma unroll 4
    for (int ci0 = 0; ci0 < C_; ci0 += 4){
      v2f av = *(const v2f*)(Abase + ci0);
      v2f bv = *(const v2f*)(Brow + ci0);
      acc = __builtin_amdgcn_wmma_f32_16x16x4_f32(false, av, false, bv,
                                                  (short)0, acc, false, false);
    }
  }
  int r0 = (tm<<4) + ((lane<16)?0:8);
  float* Yo = Y + (size_t)r0*C_ + (tn<<4) + (lane&15);
  #pragma unroll
  for (int i=0;i<8;++i) Yo[(size_t)i*C_] = acc[i];
}

// ---------------- causal depthwise conv (k=4) + SiLU, optional sequence flip -------
__global__ void k_dwconv_silu(const float* __restrict__ xz, const float* __restrict__ cw,
                              const float* __restrict__ cb, float* __restrict__ xc, int flip){
  int i = blockIdx.x*blockDim.x + threadIdx.x;
  if (i >= M_*DI_) return;
  int d = i % DI_; int m = i / DI_; int b = m / L_; int l = m % L_;
  float acc = cb[d];
  #pragma unroll
  for (int j=0;j<4;++j){
    int t = l - 3 + j;
    if (t >= 0){
      int tl = flip ? (L_-1-t) : t;
      acc += cw[d*4+j]*xz[((size_t)b*L_+tl)*384 + d];
    }
  }
  xc[(size_t)m*DI_ + d] = acc*sigm_(acc);
}

// ---------------- dt = softplus(dbl[:,:6] @ dt_w^T + dt_b) -------------------------
__global__ void k_dt(const float* __restrict__ dbl, const float* __restrict__ dtw,
                     const float* __restrict__ dtb, float* __restrict__ dt){
  int i = blockIdx.x*blockDim.x + threadIdx.x;
  if (i >= M_*DI_) return;
  int d = i % DI_; int m = i / DI_;
  float s = dtb[d];
  #pragma unroll
  for (int r=0;r<DTR_;++r) s += dbl[(size_t)m*48+r]*dtw[d*DTR_+r];
  float sp = (s>0.f) ? (s + log1pf(expf(-s))) : log1pf(expf(s));
  dt[(size_t)m*DI_+d] = sp;
}

// ---------------- selective scan: thread per (b,d), 16 states in registers ---------
__global__ void k_scan(const float* __restrict__ xc, const float* __restrict__ dt,
                       const float* __restrict__ dbl, const float* __restrict__ alog,
                       const float* __restrict__ Dp, const float* __restrict__ xz,
                       float* __restrict__ ys, int flip, int accum){
  int d = threadIdx.x; int b = blockIdx.x;
  if (d >= DI_) return;
  float Av[DS_], hs[DS_];
  #pragma unroll
  for (int n=0;n<DS_;++n){ Av[n] = -expf(alog[d*DS_+n]); hs[n] = 0.f; }
  float Dd = Dp[d];
  for (int l=0;l<L_;++l){
    size_t m = (size_t)b*L_ + l;
    float delta = dt[m*DI_+d];
    float u     = xc[m*DI_+d];
    float du = delta*u;
    float acc = 0.f;
    const float* row = dbl + m*48;
    #pragma unroll
    for (int n=0;n<DS_;++n){
      float dA = expf(delta*Av[n]);
      hs[n] = hs[n]*dA + du*row[6+n];
      acc += hs[n]*row[22+n];
    }
    float y = acc + u*Dd;
    int lo = flip ? (L_-1-l) : l;
    size_t mo = (size_t)b*L_ + lo;
    float zv = xz[mo*384 + DI_ + d];
    float out = y * (zv*sigm_(zv));
    if (accum) ys[mo*DI_+d] += out; else ys[mo*DI_+d] = out;
  }
}

// ---------------- batchnorm stats over (B,H,W) per channel -------------------------
__global__ void k_bn_stats(const float* __restrict__ y, float* __restrict__ mean,
                           float* __restrict__ var){
  __shared__ float ss[256], ss2[256];
  int ch = blockIdx.x; int t = threadIdx.x;
  float s=0.f, s2=0.f;
  for (int m=t; m<M_; m+=256){ float v = y[(size_t)m*C_+ch]; s+=v; s2+=v*v; }
  ss[t]=s; ss2[t]=s2; __syncthreads();
  for (int st=128; st>0; st>>=1){
    if (t<st){ ss[t]+=ss[t+st]; ss2[t]+=ss2[t+st]; } __syncthreads();
  }
  if (t==0){ float mu = ss[0]/(float)M_; mean[ch]=mu; var[ch]=ss2[0]/(float)M_-mu*mu; }
}

__global__ void k_bn_relu(float* __restrict__ y, const float* __restrict__ g,
                          const float* __restrict__ bt, const float* __restrict__ mean,
                          const float* __restrict__ var){
  int i = blockIdx.x*blockDim.x + threadIdx.x;
  if (i >= M_*C_) return;
  int n = i % C_;
  float v = (y[i]-mean[n])*rsqrtf(var[n]+1e-5f)*g[n]+bt[n];
  y[i] = fmaxf(v, 0.f);
}

// f_grad = relu(bn2(conv2) + f_high) -> NCHW ; f_high recomputed inline
__global__ void k_bn_res_relu(const float* __restrict__ y, const float* __restrict__ g,
                              const float* __restrict__ bt, const float* __restrict__ mean,
                              const float* __restrict__ var, const float* __restrict__ x,
                              const float* __restrict__ gru, const float* __restrict__ alpha,
                              float* __restrict__ fg){
  int i = blockIdx.x*blockDim.x + threadIdx.x;
  if (i >= M_*C_) return;
  int n = i % C_; int m = i / C_; int b = m/L_; int p = m%L_;
  float v = (y[i]-mean[n])*rsqrtf(var[n]+1e-5f)*g[n]+bt[n];
  size_t o = ((size_t)b*C_+n)*HW_ + p;
  float fh = x[o]*(1.f + alpha[0]*gru[b*HW_+p]);
  fg[o] = fmaxf(v + fh, 0.f);
}

__global__ void k_add(const float* __restrict__ a, const float* __restrict__ b,
                      float* __restrict__ c, int n){
  int i = blockIdx.x*blockDim.x + threadIdx.x;
  if (i < n) c[i] = a[i] + b[i];
}

__global__ void k_ga(const float* __restrict__ xa, float* __restrict__ ga){
  __shared__ float ss[256];
  int bc = blockIdx.x; int t = threadIdx.x;
  const float* p = xa + (size_t)bc*HW_;
  float s=0.f;
  for (int i=t; i<HW_; i+=256) s += p[i];
  ss[t]=s; __syncthreads();
  for (int st=128; st>0; st>>=1){ if (t<st) ss[t]+=ss[t+st]; __syncthreads(); }
  if (t==0) ga[bc] = ss[0]/(float)HW_;
}

__global__ void k_xgl(const float* __restrict__ ga, const float* __restrict__ w1,
                      const float* __restrict__ b1, const float* __restrict__ w2,
                      const float* __restrict__ b2, float* __restrict__ xgl){
  int t = blockIdx.x*blockDim.x + threadIdx.x;
  if (t >= B_*C_) return;
  int b = t/C_, co = t%C_;
  float acc = b2[co];
  for (int r=0;r<RI_;++r){
    float s = b1[r];
    for (int c=0;c<C_;++c) s += ga[b*C_+c]*w1[r*C_+c];
    acc += lrelu_(s)*w2[co*RI_+r];
  }
  xgl[t] = acc;
}

// local attention 1x1 convs + blend -> f_out
__global__ void k_final(const float* __restrict__ xa, const float* __restrict__ w1,
                        const float* __restrict__ b1, const float* __restrict__ w2,
                        const float* __restrict__ b2, const float* __restrict__ xgl,
                        const float* __restrict__ fi, const float* __restrict__ fg,
                        float* __restrict__ fo){
  int i = blockIdx.x*blockDim.x + threadIdx.x;
  if (i >= B_*HW_) return;
  int b = i/HW_, p = i%HW_;
  float t[RI_];
  for (int r=0;r<RI_;++r){
    float s = b1[r];
    for (int c=0;c<C_;++c) s += xa[((size_t)b*C_+c)*HW_+p]*w1[r*C_+c];
    t[r] = lrelu_(s);
  }
  for (int c=0;c<C_;++c){
    float xl = b2[c];
    #pragma unroll
    for (int r=0;r<RI_;++r) xl += t[r]*w2[c*RI_+r];
    float wv = sigm_(xl + xgl[b*C_+c]);
    size_t o = ((size_t)b*C_+c)*HW_ + p;
    fo[o] = 2.f*fi[o]*wv + 2.f*fg[o]*(1.f-wv);
  }
}

extern "C" void kernel_launch(void* const* d_in, const int* in_sizes, int n_in,
                              void* d_out, int out_size, void* d_ws, size_t ws_size,
                              hipStream_t stream) {
  (void)in_sizes; (void)n_in; (void)out_size; (void)ws_size;
  // pytree order: x, illumination, gradient, then params sorted alphabetically
  const float* x_in  = (const float*)d_in[0];
  const float* illum = (const float*)d_in[1];
  const float* grad  = (const float*)d_in[2];
  const float* alpha   = (const float*)d_in[3];
  const float* blk_lnb = (const float*)d_in[4];
  const float* blk_lnw = (const float*)d_in[5];
  const float* bn1_b = (const float*)d_in[6];
  const float* bn1_g = (const float*)d_in[7];
  const float* bn2_b = (const float*)d_in[8];
  const float* bn2_g = (const float*)d_in[9];
  const float* bA = (const float*)d_in[10]; const float* bD = (const float*)d_in[11];
  const float* bcb = (const float*)d_in[12]; const float* bcw = (const float*)d_in[13];
  const float* bdtb = (const float*)d_in[14]; const float* bdtw = (const float*)d_in[15];
  const float* bxp = (const float*)d_in[16];
  const float* cnn_w1 = (const float*)d_in[17];
  const float* cnn_w2 = (const float*)d_in[18];
  const float* fA = (const float*)d_in[19]; const float* fD = (const float*)d_in[20];
  const float* fcb = (const float*)d_in[21]; const float* fcw = (const float*)d_in[22];
  const float* fdtb = (const float*)d_in[23]; const float* fdtw = (const float*)d_in[24];
  const float* fxp = (const float*)d_in[25];
  const float* ga_b1 = (const float*)d_in[26]; const float* ga_b2 = (const float*)d_in[27];
  const float* ga_w1 = (const float*)d_in[28]; const float* ga_w2 = (const float*)d_in[29];
  const float* ic_b = (const float*)d_in[30]; const float* ic_w = (const float*)d_in[31];
  const float* in_proj = (const float*)d_in[32];
  const float* la_b1 = (const float*)d_in[33]; const float* la_b2 = (const float*)d_in[34];
  const float* la_w1 = (const float*)d_in[35]; const float* la_w2 = (const float*)d_in[36];
  const float* ln_b = (const float*)d_in[37]; const float* ln_w = (const float*)d_in[38];
  const float* out_proj = (const float*)d_in[39];

  float* out0 = (float*)d_out;           // f_out
  float* out1 = out0 + OUTN_;            // f_illum
  float* out2 = out0 + 2*OUTN_;          // f_grad

  // workspace carve-up (floats)
  float* ws = (float*)d_ws; size_t off = 0;
  float* ILU = ws + off; off += (size_t)B_*HW_;
  float* GRU = ws + off; off += (size_t)B_*HW_;
  float* FLOW= ws + off; off += OUTN_;                 // f_low NCHW; later conv1 out (M,96)
  float* SEQ = ws + off; off += (size_t)M_*C_;         // seq; later xg_pre; later xa NCHW
  float* HB  = ws + off; off += (size_t)M_*C_;         // layernorm output h
  float* XZ  = ws + off; off += (size_t)M_*384;
  float* XC  = ws + off; off += (size_t)M_*DI_;        // xc; later conv2 out (M,96)
  float* DTB = ws + off; off += (size_t)M_*DI_;
  float* YS  = ws + off; off += (size_t)M_*DI_;
  float* DBL = ws + off; off += (size_t)M_*48;
  float* PADP= ws + off; off += (size_t)B_*PHW_*C_;    // padded pixel-major conv input
  float* WXP = ws + off; off += (size_t)48*DI_;        // zero-padded x_proj weights
  float* ST  = ws + off; off += 4*C_;                  // bn1 mean/var, bn2 mean/var
  float* GA  = ws + off; off += (size_t)B_*C_;
  float* XGL = ws + off; off += (size_t)B_*C_;

  const int T = 256;
  const int TM8 = (M_/16)/8;   // 144 blocks of 8 M-tiles

  // stage A: upsample, mask, box+transpose
  k_upsample2x<<<(B_*HW_+T-1)/T, T, 0, stream>>>(illum, ILU);
  k_upsample2x<<<(B_*HW_+T-1)/T, T, 0, stream>>>(grad,  GRU);
  k_mask_flow<<<(B_*HW_+T-1)/T, T, 0, stream>>>(x_in, ILU, ic_w, ic_b, FLOW);
  k_box_seq<<<((int)OUTN_+T-1)/T, T, 0, stream>>>(FLOW, SEQ);

  // stage B: bimamba
  k_layernorm<<<(M_+T-1)/T, T, 0, stream>>>(SEQ, blk_lnw, blk_lnb, HB);
  k_wmma_gemm_lds<96><<<24*TM8, T, 0, stream>>>(HB, in_proj, XZ, 24, 384);
  for (int dir=0; dir<2; ++dir){
    const float* Al = dir? bA  : fA;  const float* Dp = dir? bD  : fD;
    const float* cb = dir? bcb : fcb; const float* cw = dir? bcw : fcw;
    const float* db = dir? bdtb: fdtb;const float* dw = dir? bdtw: fdtw;
    const float* xp = dir? bxp : fxp;
    int flip = dir;
    k_dwconv_silu<<<(M_*DI_+T-1)/T, T, 0, stream>>>(XZ, cw, cb, XC, flip);
    k_padw<<<(48*DI_+T-1)/T, T, 0, stream>>>(xp, WXP);
    k_wmma_gemm_lds<192><<<3*TM8, T, 0, stream>>>(XC, WXP, DBL, 3, 48);
    k_dt<<<(M_*DI_+T-1)/T, T, 0, stream>>>(DBL, dw, db, DTB);
    k_scan<<<B_, DI_, 0, stream>>>(XC, DTB, DBL, Al, Dp, XZ, YS, flip, /*accum=*/dir);
  }
  k_wmma_gemm_lds<192><<<6*TM8, T, 0, stream>>>(YS, out_proj, SEQ, 6, 96);
  k_layernorm_t<<<(M_+T-1)/T, T, 0, stream>>>(SEQ, ln_w, ln_b, out1);   // f_illum

  // stage C: gradient CNN branch (both convs pixel-major, WMMA implicit GEMM)
  k_pad_fhigh_pm<<<((int)(B_*PHW_*C_)+T-1)/T, T, 0, stream>>>(x_in, GRU, alpha, PADP);
  k_wmma_conv3x3_pm<<<6*TM8, T, 0, stream>>>(PADP, cnn_w1, FLOW);
  k_bn_stats<<<C_, T, 0, stream>>>(FLOW, ST+0, ST+C_);
  k_bn_relu<<<(M_*C_+T-1)/T, T, 0, stream>>>(FLOW, bn1_g, bn1_b, ST+0, ST+C_);
  k_pad_pm<<<((int)(B_*PHW_*C_)+T-1)/T, T, 0, stream>>>(FLOW, PADP);
  k_wmma_conv3x3_pm<<<6*TM8, T, 0, stream>>>(PADP, cnn_w2, XC);
  k_bn_stats<<<C_, T, 0, stream>>>(XC, ST+2*C_, ST+3*C_);
  k_bn_res_relu<<<(M_*C_+T-1)/T, T, 0, stream>>>(XC, bn2_g, bn2_b, ST+2*C_, ST+3*C_,
                                                 x_in, GRU, alpha, out2);

  // stage D: fusion attention
  k_add<<<((int)OUTN_+T-1)/T, T, 0, stream>>>(out1, out2, SEQ, (int)OUTN_); // xa NCHW
  k_ga<<<B_*C_, T, 0, stream>>>(SEQ, GA);
  k_xgl<<<1, T, 0, stream>>>(GA, ga_w1, ga_b1, ga_w2, ga_b2, XGL);
  k_final<<<(B_*HW_+T-1)/T, T, 0, stream>>>(SEQ, la_w1, la_b1, la_w2, la_b2, XGL,
                                            out1, out2, out0);
}